// LocalModel_10763188043965
// MI455X (gfx1250) — compile-verified
//
#include <hip/hip_runtime.h>
#include <hip/hip_bf16.h>

typedef __attribute__((ext_vector_type(2))) float v2f;
typedef __attribute__((ext_vector_type(8))) float v8f;

#define GN 100000      // nodes
#define GE 1600000     // edges
#define GD 128         // feature dim
#define BN_EPS 1e-5f
#define STATS_BLOCKS 500
#define ROWS_PER_STATS (GN / STATS_BLOCKS)   // 200

// ---------------------------------------------------------------------------
// 1) deg = 1.0 (self loop), sums[256] = 0.  ws layout: [deg: N][sums: 256][h: N*D]
// ---------------------------------------------------------------------------
__global__ void init_deg_sums(float* __restrict__ ws) {
    int i = blockIdx.x * blockDim.x + threadIdx.x;
    if (i < GN) ws[i] = 1.0f;          // deg starts at 1 (self loop)
    else if (i < GN + 256) ws[i] = 0.0f;  // sum / sumsq accumulators
}

// ---------------------------------------------------------------------------
// 2) degree histogram over dst, f32 native atomics
// ---------------------------------------------------------------------------
__global__ void degree_kernel(const int* __restrict__ ei, float* __restrict__ deg) {
    int e = blockIdx.x * blockDim.x + threadIdx.x;
    if (e < GE) {
        int d = ei[GE + e];            // dst row of edge_index
        unsafeAtomicAdd(&deg[d], 1.0f);
    }
}

// 3) deg <- rsqrt(deg)   (deg >= 1 always, matches reference's where(deg>0))
__global__ void rsqrt_kernel(float* __restrict__ deg) {
    int i = blockIdx.x * blockDim.x + threadIdx.x;
    if (i < GN) deg[i] = __frsqrt_rn(deg[i]);
}

// ---------------------------------------------------------------------------
// 4) h = x @ W  via V_WMMA_F32_16X16X4_F32 (exact fp32).
//    Block = 256 threads = 8 waves; wave w computes tile (mBase=16*blockIdx.x,
//    nBase=16*w). K-loop: 128/4 = 32 wmma ops per tile.
//    A 16x4 layout: lanes 0-15 M=lane (K=0,1 in v[0],v[1]); lanes 16-31 K=2,3.
//    B 4x16 layout: v[0] lanes0-15 K=0 row, lanes16-31 K=2; v[1]: K=1 / K=3.
//    C/D 16x16: VGPR r -> M=r (lanes 0-15), M=r+8 (lanes 16-31), N=lane%16.
// ---------------------------------------------------------------------------
__global__ void __launch_bounds__(256)
gemm_wmma_kernel(const float* __restrict__ x, const float* __restrict__ W,
                 float* __restrict__ h) {
    const int wv   = threadIdx.x >> 5;        // 0..7 -> column tile
    const int ln   = threadIdx.x & 31;
    const int l16  = ln & 15;
    const int koff = (ln >> 4) << 1;          // 0 for lanes 0-15, 2 for 16-31
    const int mBase = blockIdx.x << 4;
    const int nBase = wv << 4;

    const float* __restrict__ xrow = x + (size_t)(mBase + l16) * GD;

    v8f c = {};
#pragma unroll 4
    for (int k0 = 0; k0 < GD; k0 += 4) {
        v2f a, b;
        a.x = xrow[k0 + koff];
        a.y = xrow[k0 + koff + 1];
        b.x = W[(size_t)(k0 + koff) * GD + nBase + l16];
        b.y = W[(size_t)(k0 + koff + 1) * GD + nBase + l16];
        c = __builtin_amdgcn_wmma_f32_16x16x4_f32(
                /*neg_a=*/false, a, /*neg_b=*/false, b,
                /*c_mod=*/(short)0, c, /*reuse_a=*/false, /*reuse_b=*/false);
    }

    const int col    = nBase + l16;
    const int rowTop = mBase + ((ln >> 4) << 3);   // +8 for lanes 16-31
#pragma unroll
    for (int r = 0; r < 8; ++r)
        h[(size_t)(rowTop + r) * GD + col] = c[r];
}

// ---------------------------------------------------------------------------
// 5) accumulator seed (in d_out): out = x + b + h * dinv^2   (self-loop+bias+residual)
// ---------------------------------------------------------------------------
__global__ void init_hl_kernel(const float* __restrict__ x, const float* __restrict__ h,
                               const float* __restrict__ dinv, const float* __restrict__ bias,
                               float* __restrict__ out) {
    int i = blockIdx.x * blockDim.x + threadIdx.x;   // < N*D
    int row = i >> 7;
    int d   = i & (GD - 1);
    float di = dinv[row];
    out[i] = x[i] + bias[d] + h[i] * di * di;
}

// ---------------------------------------------------------------------------
// 6) edge scatter: one wave per edge; 32 lanes x float4 = full 128-dim row.
//    out[dst] += h[src] * dinv[src]*dinv[dst]
// ---------------------------------------------------------------------------
__global__ void __launch_bounds__(256)
edge_agg_kernel(const int* __restrict__ ei, const float* __restrict__ h,
                const float* __restrict__ dinv, float* __restrict__ out) {
    int gid  = blockIdx.x * blockDim.x + threadIdx.x;
    int e    = gid >> 5;
    int ln   = gid & 31;
    if (e >= GE) return;
    int s = ei[e];
    int d = ei[GE + e];
    float norm = dinv[s] * dinv[d];

    const float4* __restrict__ hs = (const float4*)(h + (size_t)s * GD);
    float4 v = hs[ln];                       // b128 gather
    float* dstp = out + (size_t)d * GD + (ln << 2);
    unsafeAtomicAdd(dstp + 0, v.x * norm);   // native global_atomic_add_f32
    unsafeAtomicAdd(dstp + 1, v.y * norm);
    unsafeAtomicAdd(dstp + 2, v.z * norm);
    unsafeAtomicAdd(dstp + 3, v.w * norm);
}

// ---------------------------------------------------------------------------
// 7) BN stats: thread d of each block accumulates sum/sumsq for feature d over
//    a row stripe (coalesced), then 2 atomics per thread.
// ---------------------------------------------------------------------------
__global__ void __launch_bounds__(GD)
bn_stats_kernel(const float* __restrict__ hl, float* __restrict__ sums) {
    int d  = threadIdx.x;
    int r0 = blockIdx.x * ROWS_PER_STATS;
    float s = 0.0f, sq = 0.0f;
    const float* __restrict__ p = hl + (size_t)r0 * GD + d;
    for (int r = 0; r < ROWS_PER_STATS; ++r) {
        float v = p[(size_t)r * GD];
        s  += v;
        sq += v * v;
    }
    unsafeAtomicAdd(&sums[d], s);
    unsafeAtomicAdd(&sums[GD + d], sq);
}

// ---------------------------------------------------------------------------
// 8) BN apply in place: out = gamma*(v-mean)*rsqrt(var+eps)+beta; biased var.
// ---------------------------------------------------------------------------
__global__ void bn_apply_kernel(float* __restrict__ out, const float* __restrict__ sums,
                                const float* __restrict__ gamma, const float* __restrict__ beta) {
    int i = blockIdx.x * blockDim.x + threadIdx.x;   // < N*D
    int d = i & (GD - 1);
    const float invN = 1.0f / (float)GN;
    float mean = sums[d] * invN;
    float var  = sums[GD + d] * invN - mean * mean;
    float v    = out[i];
    out[i] = gamma[d] * (v - mean) * __frsqrt_rn(var + BN_EPS) + beta[d];
}

// ---------------------------------------------------------------------------
extern "C" void kernel_launch(void* const* d_in, const int* in_sizes, int n_in,
                              void* d_out, int out_size, void* d_ws, size_t ws_size,
                              hipStream_t stream) {
    const float* x     = (const float*)d_in[0];
    const int*   ei    = (const int*)  d_in[1];
    // d_in[2] virt_h, d_in[3] virt_edge_index: unused by the reference
    const float* W     = (const float*)d_in[4];
    const float* bias  = (const float*)d_in[5];
    const float* gamma = (const float*)d_in[6];
    const float* beta  = (const float*)d_in[7];
    float* out = (float*)d_out;

    // workspace layout (floats): [deg/dinv : N][sums : 256][h : N*D]
    float* ws   = (float*)d_ws;
    float* deg  = ws;                 // becomes dinv after rsqrt_kernel
    float* sums = ws + GN;
    float* h    = ws + GN + 256;

    const int NTOT = GN * GD;

    // 1) init deg=1, sums=0
    init_deg_sums<<<(GN + 256 + 255) / 256, 256, 0, stream>>>(ws);
    // 2) degree histogram
    degree_kernel<<<(GE + 255) / 256, 256, 0, stream>>>(ei, deg);
    // 3) dinv = rsqrt(deg)
    rsqrt_kernel<<<(GN + 255) / 256, 256, 0, stream>>>(deg);
    // 4) h = x @ W   (WMMA f32, 6250 blocks x 8 waves)
    gemm_wmma_kernel<<<GN / 16, 256, 0, stream>>>(x, W, h);
    // 5) out = x + b + h*dinv^2
    init_hl_kernel<<<NTOT / 256, 256, 0, stream>>>(x, h, deg, bias, out);
    // 6) out[dst] += h[src]*norm  (1 wave/edge)
    edge_agg_kernel<<<(GE + 7) / 8, 256, 0, stream>>>(ei, h, deg, out);
    // 7) per-feature sum / sumsq
    bn_stats_kernel<<<STATS_BLOCKS, GD, 0, stream>>>(out, sums);
    // 8) batchnorm apply in place
    bn_apply_kernel<<<NTOT / 256, 256, 0, stream>>>(out, sums, gamma, beta);
}